// ChamferLoss_28595892257476
// MI455X (gfx1250) — compile-verified
//
#include <hip/hip_runtime.h>

typedef float v2f __attribute__((ext_vector_type(2)));
typedef float v8f __attribute__((ext_vector_type(8)));

#define WAVES_PER_BLOCK 8
#define ROWS_PER_BLOCK (WAVES_PER_BLOCK * 16)
#define CHUNK 2048               // columns staged per LDS pass (32 KB)

// For each of `nrows` points in P (per batch), compute min squared distance to
// all `ncols` points in Q via the augmented K=4 GEMM on V_WMMA_F32_16X16X4_F32:
//   dist = [x0, x1, |x|^2, 1] . [-2y0, -2y1, 1, |y|^2]
// The augmented B matrix is staged in LDS once per block (amortized across the
// 8 waves), so the steady-state tile loop is ds_load_b64 + wmma + min3 only.
__global__ __launch_bounds__(256) void chamfer_min_kernel(
    const float2* __restrict__ P, const float2* __restrict__ Q,
    float* __restrict__ minOut, int nrows, int ncols)
{
    __shared__ float4 baug[CHUNK];   // per column: [-2qx, -2qy, 1, |q|^2]

    const int batch = blockIdx.y;
    const int tid   = threadIdx.x;
    const int lane  = tid & 31;
    const int wave  = tid >> 5;
    const int half  = lane >> 4;     // 0: lanes 0-15 (K={0,1}), 1: lanes 16-31 (K={2,3})
    const int sub   = lane & 15;

    const int rowBase = (blockIdx.x * WAVES_PER_BLOCK + wave) * 16;
    const float2* __restrict__ Pb = P + (size_t)batch * nrows;
    const float2* __restrict__ Qb = Q + (size_t)batch * ncols;

    // Loop-invariant A fragment (16x4 f32), rows rowBase..rowBase+15.
    float2 p = Pb[rowBase + sub];
    float pn = p.x * p.x + p.y * p.y;
    v2f a;
    a.x = half ? pn   : p.x;
    a.y = half ? 1.0f : p.y;

    v8f acc;
#pragma unroll
    for (int r = 0; r < 8; ++r) acc[r] = 3.0e38f;

    for (int cb = 0; cb < ncols; cb += CHUNK) {
        // Cooperative fill: 256 threads x (CHUNK/256) columns each.
#pragma unroll
        for (int i = 0; i < CHUNK / 256; ++i) {
            int c = i * 256 + tid;
            float2 q = Qb[cb + c];
            baug[c] = make_float4(-2.0f * q.x, -2.0f * q.y, 1.0f,
                                  q.x * q.x + q.y * q.y);
        }
        __syncthreads();

        // Consume: each wave runs all CHUNK/16 tiles against its A strip.
        const float2* __restrict__ bp = (const float2*)baug;  // 2 x float2 per column
#pragma unroll 4
        for (int t = 0; t < CHUNK / 16; ++t) {
            // B fragment (4x16 f32): lane half selects K={0,1} vs K={2,3}.
            float2 bq = bp[(t * 16 + sub) * 2 + half];
            v2f b;
            b.x = bq.x;
            b.y = bq.y;

            v8f c = {};  // fresh accumulator: D = full distance tile
            // 8 args: (neg_a, A, neg_b, B, c_mod, C, reuse_a, reuse_b)
            v8f d = __builtin_amdgcn_wmma_f32_16x16x4_f32(
                false, a, false, b, (short)0, c, false, false);

#pragma unroll
            for (int r = 0; r < 8; ++r) acc[r] = fminf(acc[r], d[r]);
        }
        __syncthreads();
    }

    // D layout: slot r, lane<16 -> (M=r, N=lane); lane>=16 -> (M=r+8, N=lane-16).
    // Min across the 16 lanes of each half yields 8 row-mins per half.
#pragma unroll
    for (int m = 1; m <= 8; m <<= 1) {
#pragma unroll
        for (int r = 0; r < 8; ++r)
            acc[r] = fminf(acc[r], __shfl_xor(acc[r], m, 32));
    }

    if (sub == 0) {
        float* outp = minOut + (size_t)batch * nrows + rowBase + (half ? 8 : 0);
#pragma unroll
        for (int r = 0; r < 8; ++r) outp[r] = acc[r];
    }
}

// Sum the two min arrays and emit mean(rowmins) + mean(colmins).
__global__ __launch_bounds__(256) void chamfer_finalize_kernel(
    const float* __restrict__ mins, int n1, int n2, float* __restrict__ out)
{
    __shared__ float s1[256];
    __shared__ float s2[256];
    float sum1 = 0.0f, sum2 = 0.0f;
    for (int i = threadIdx.x; i < n1; i += 256) sum1 += mins[i];
    for (int i = threadIdx.x; i < n2; i += 256) sum2 += mins[n1 + i];
    s1[threadIdx.x] = sum1;
    s2[threadIdx.x] = sum2;
    __syncthreads();
    for (int s = 128; s > 0; s >>= 1) {
        if (threadIdx.x < s) {
            s1[threadIdx.x] += s1[threadIdx.x + s];
            s2[threadIdx.x] += s2[threadIdx.x + s];
        }
        __syncthreads();
    }
    if (threadIdx.x == 0)
        out[0] = s1[0] / (float)n1 + s2[0] / (float)n2;
}

extern "C" void kernel_launch(void* const* d_in, const int* in_sizes, int n_in,
                              void* d_out, int out_size, void* d_ws, size_t ws_size,
                              hipStream_t stream) {
    (void)n_in; (void)out_size; (void)ws_size;
    const float2* pred = (const float2*)d_in[0];  // [B, N, 2] fp32
    const float2* gt   = (const float2*)d_in[1];  // [B, M, 2] fp32

    const int B = 4;
    const int N = in_sizes[0] / (B * 2);
    const int M = in_sizes[1] / (B * 2);

    float* rowmin = (float*)d_ws;              // B*N floats: min over gt per pred
    float* colmin = rowmin + (size_t)B * N;    // B*M floats: min over pred per gt

    dim3 block(256);
    // Pass 1: per pred point, min distance to gt points.
    chamfer_min_kernel<<<dim3(N / ROWS_PER_BLOCK, B), block, 0, stream>>>(
        pred, gt, rowmin, N, M);
    // Pass 2: per gt point, min distance to pred points (symmetric).
    chamfer_min_kernel<<<dim3(M / ROWS_PER_BLOCK, B), block, 0, stream>>>(
        gt, pred, colmin, M, N);
    // Final scalar.
    chamfer_finalize_kernel<<<1, block, 0, stream>>>(
        rowmin, B * N, B * M, (float*)d_out);
}